// TokenSelection_67130338836483
// MI455X (gfx1250) — compile-verified
//
#include <hip/hip_runtime.h>

// ---------------- problem constants (from setup_inputs) ----------------
#define NBATCH 2
#define NHEADS 8
#define NTOK   16384          // N
#define NC     256            // dim / channels
#define NCMP   256            // number of candidate blocks
#define NSEL   64             // num_select
#define BSZ    8              // spatial block size
#define HGRID  16             // 128/8 blocks per row
#define NROWS  (NHEADS * NTOK)     // 131072 reduction rows per batch
#define NCHUNK 256                 // reduction chunks per batch
#define ROWS_PER_CHUNK (NROWS / NCHUNK)   // 512

#define KSLC_ELEMS ((size_t)NBATCH * NSEL * BSZ * BSZ * NC)   // 2,097,152

typedef int v4i __attribute__((vector_size(16)));
typedef __attribute__((address_space(1))) v4i g_v4i;   // global int4
typedef __attribute__((address_space(3))) v4i l_v4i;   // LDS int4

// ---------------------------------------------------------------------
// Kernel A: per-chunk column sums of attn[b, h, n, c] over (h,n), f64 acc.
// Fully coalesced 1KB rows, non-temporal (256MB streamed exactly once).
// ---------------------------------------------------------------------
__global__ void importance_partials_kernel(const float* __restrict__ attn,
                                           double* __restrict__ partials) {
    const int c     = threadIdx.x;   // 0..255 channel
    const int chunk = blockIdx.x;    // 0..NCHUNK-1
    const int b     = blockIdx.y;    // batch
    size_t base = ((size_t)b * NROWS + (size_t)chunk * ROWS_PER_CHUNK) * NC + c;
    double acc = 0.0;
#pragma unroll 4
    for (int r = 0; r < ROWS_PER_CHUNK; ++r) {
        acc += (double)__builtin_nontemporal_load(attn + base + (size_t)r * NC);
    }
    partials[((size_t)b * NCHUNK + chunk) * NC + c] = acc;
}

// ---------------------------------------------------------------------
// Kernel B: finish the reduction in a fixed order (deterministic), then
// stable-descending top-64 via rank counting (ties -> lower index first,
// matching jax.lax.top_k). Writes int indices to ws and float copies to
// the tail of d_out.
// ---------------------------------------------------------------------
__global__ void topk_select_kernel(const double* __restrict__ partials,
                                   int* __restrict__ selWs,
                                   float* __restrict__ outIdx) {
    __shared__ double simp[NCMP];
    __shared__ int    sidx[NSEL];
    const int c = threadIdx.x;   // candidate block id 0..255
    const int b = blockIdx.x;
    double acc = 0.0;
    for (int ch = 0; ch < NCHUNK; ++ch)
        acc += partials[((size_t)b * NCHUNK + ch) * NC + c];
    simp[c] = acc;               // (scale by 1/N irrelevant for ordering)
    __syncthreads();
    const double mv = simp[c];
    int rank = 0;
    for (int j = 0; j < NCMP; ++j) {
        double o = simp[j];
        rank += (o > mv) || (o == mv && j < c);
    }
    if (rank < NSEL) sidx[rank] = c;
    __syncthreads();
    if (c < NSEL) {
        selWs[b * NSEL + c]  = sidx[c];
        outIdx[b * NSEL + c] = (float)sidx[c];
    }
}

// ---------------------------------------------------------------------
// Kernel C: gather + transpose one selected 8x8 block of K or V.
//   out[b][sel][o] with o = c*64 + m ;  src = X[b, n_src(m), c]
//   n_src(m) = (bh*8 + m/8)*128 + bw*8 + (m%8)
// Stage a 32-row half tile (32 x 256 f32, stride 257) in LDS using
// CDNA5 async global->LDS B128 copies (ASYNCcnt), then emit transposed
// float4 stores (conflict-free LDS column reads, 128B store runs).
// ---------------------------------------------------------------------
__global__ void gather_transpose_kernel(const float* __restrict__ kIn,
                                        const float* __restrict__ vIn,
                                        const int* __restrict__ selWs,
                                        float* __restrict__ out) {
    __shared__ float tile[32][257];
    const int tid = threadIdx.x;      // 0..255
    const int sel = blockIdx.x;       // 0..63
    const int b   = blockIdx.y;       // 0..1
    const int z   = blockIdx.z;       // 0 = K, 1 = V
    const float* src = z ? vIn : kIn;
    const int blk    = selWs[b * NSEL + sel];
    const int base_n = (blk >> 4) * (BSZ * 128) + (blk & (HGRID - 1)) * BSZ;
    float* chunkOut = out + (size_t)z * KSLC_ELEMS
                          + ((size_t)b * NSEL + sel) * (size_t)(BSZ * BSZ * NC);

    for (int half = 0; half < 2; ++half) {
        // --- async load: 32 rows x 256 ch = 2048 x B128, coalesced 512B/wave
#pragma unroll
        for (int it = 0; it < 8; ++it) {
            int e    = it * 256 + tid;
            int c4   = e & 63;        // float4 group along channels
            int mloc = e >> 6;        // 0..31 row within half tile
            int m    = half * 32 + mloc;
            int n_src = base_n + (m >> 3) * 128 + (m & 7);
            const float* g = src + (((size_t)b * NTOK + n_src) * NC + c4 * 4);
            __builtin_amdgcn_global_load_async_to_lds_b128(
                (g_v4i*)g, (l_v4i*)&tile[mloc][c4 * 4], 0, 0);
        }
        asm volatile("s_wait_asynccnt 0" ::: "memory");
        __syncthreads();
        // --- transposed store: out[c][m], float4 along m
#pragma unroll
        for (int it = 0; it < 8; ++it) {
            int e  = it * 256 + tid;
            int m4 = e & 7;           // float4 group along m within half
            int c  = e >> 3;          // 0..255
            float4 val;
            val.x = tile[m4 * 4 + 0][c];
            val.y = tile[m4 * 4 + 1][c];
            val.z = tile[m4 * 4 + 2][c];
            val.w = tile[m4 * 4 + 3][c];
            *(float4*)(chunkOut + c * 64 + half * 32 + m4 * 4) = val;
        }
        __syncthreads();
    }
}

// ---------------------------------------------------------------------
extern "C" void kernel_launch(void* const* d_in, const int* in_sizes, int n_in,
                              void* d_out, int out_size, void* d_ws, size_t ws_size,
                              hipStream_t stream) {
    // inputs: 0=q (unused), 1=k, 2=v, 3=attn_scores_cmp, 4=spatial_h, 5=spatial_w
    const float* kIn  = (const float*)d_in[1];
    const float* vIn  = (const float*)d_in[2];
    const float* attn = (const float*)d_in[3];
    float* out = (float*)d_out;

    double* partials = (double*)d_ws;
    int* selWs = (int*)((char*)d_ws + sizeof(double) * (size_t)NBATCH * NCHUNK * NC);

    float* outIdx = out + 2 * KSLC_ELEMS;   // indices tail of d_out (as float)

    importance_partials_kernel<<<dim3(NCHUNK, NBATCH), 256, 0, stream>>>(attn, partials);
    topk_select_kernel<<<dim3(NBATCH), 256, 0, stream>>>(partials, selWs, outIdx);
    gather_transpose_kernel<<<dim3(NSEL, NBATCH, 2), 256, 0, stream>>>(kIn, vIn, selWs, out);
}